// GAT_41446434406884
// MI455X (gfx1250) — compile-verified
//
#include <hip/hip_runtime.h>

// CDNA5 / gfx1250 GAT-style kernel:
//   score = <self, [neigh; self]> ; attn = softmax(score) ;
//   ctx = attn @ [neigh; self] ; out = relu(ctx @ W)
// Final 16x128 @ 128x128 GEMM per node-tile runs on V_WMMA_F32_16X16X4_F32
// (full f32 precision; problem is HBM-bound at ~4.5 FLOP/byte: 0.92 GB of
// traffic -> ~40us floor at 23.3 TB/s, vs ~4.1 GFLOP total).

typedef __attribute__((ext_vector_type(2))) float v2f;
typedef __attribute__((ext_vector_type(8))) float v8f;

namespace {
constexpr int KNEI = 16;          // neighbors per node
constexpr int D    = 128;         // input dim
constexpr int DOUT = 128;         // output dim
constexpr int TILE = 16;          // nodes per workgroup (M-tile of the GEMM)
constexpr int CTXS = D + 4;       // LDS row stride (pad: A-frag reads hit 16 banks)
}

__global__ __launch_bounds__(256) void gat_attn_wmma(
    const float* __restrict__ self_vecs,   // [N, D]
    const float* __restrict__ neigh_vecs,  // [N, K, D]
    const float* __restrict__ weights,     // [D, DOUT]
    float* __restrict__ out,               // [N, DOUT]
    int n_nodes)
{
  __shared__ float s_scores[TILE][KNEI + 1];  // 17 logits per node
  __shared__ float s_attn  [TILE][KNEI + 2];  // padded
  __shared__ float s_ctx   [TILE][CTXS];      // context rows (A matrix)

  const int tid  = threadIdx.x;
  const int base = blockIdx.x * TILE;

  // ---- zero score accumulators ----
  for (int i = tid; i < TILE * (KNEI + 1); i += 256)
    (&s_scores[0][0])[i] = 0.0f;
  __syncthreads();

  // ---- phase 1: attention logits -------------------------------------
  // thread (node, seg): node = tid>>4 (0..15), seg = tid&15 covers 8 floats
  // -> each 16-lane group streams 512B contiguous per row (coalesced).
  const int node = tid >> 4;
  const int seg  = tid & 15;
  int gnode = base + node;
  if (gnode >= n_nodes) gnode = n_nodes - 1;   // tail clamp (N % 16 == 0 here)

  const float* sp = self_vecs + (size_t)gnode * D + seg * 8;
  const float4 sv0 = *reinterpret_cast<const float4*>(sp);
  const float4 sv1 = *reinterpret_cast<const float4*>(sp + 4);

  const float* np_base = neigh_vecs + (size_t)gnode * KNEI * D + seg * 8;

  #pragma unroll 4
  for (int k = 0; k < KNEI; ++k) {
    const float* np = np_base + k * D;
    const float4 nv0 = *reinterpret_cast<const float4*>(np);
    const float4 nv1 = *reinterpret_cast<const float4*>(np + 4);
    float p = sv0.x * nv0.x + sv0.y * nv0.y + sv0.z * nv0.z + sv0.w * nv0.w
            + sv1.x * nv1.x + sv1.y * nv1.y + sv1.z * nv1.z + sv1.w * nv1.w;
    atomicAdd(&s_scores[node][k], p);          // ds_add_f32
  }
  {
    float p = sv0.x * sv0.x + sv0.y * sv0.y + sv0.z * sv0.z + sv0.w * sv0.w
            + sv1.x * sv1.x + sv1.y * sv1.y + sv1.z * sv1.z + sv1.w * sv1.w;
    atomicAdd(&s_scores[node][KNEI], p);       // self-as-neighbor logit
  }
  __syncthreads();

  // ---- phase 2: softmax over 17 (one thread per node) -----------------
  if (tid < TILE) {
    float sc[KNEI + 1];
    float mx = -3.402823466e38f;
    #pragma unroll
    for (int k = 0; k <= KNEI; ++k) { sc[k] = s_scores[tid][k]; mx = fmaxf(mx, sc[k]); }
    float sum = 0.0f;
    #pragma unroll
    for (int k = 0; k <= KNEI; ++k) { sc[k] = __expf(sc[k] - mx); sum += sc[k]; }
    const float inv = 1.0f / sum;
    #pragma unroll
    for (int k = 0; k <= KNEI; ++k) s_attn[tid][k] = sc[k] * inv;
  }
  __syncthreads();

  // ---- phase 3: context = attn . [neigh; self] ------------------------
  // Re-reads the same 8KB neighbor tile -> hits WGP$/L2, no extra HBM.
  {
    const float aself = s_attn[node][KNEI];
    float acc[8] = { aself * sv0.x, aself * sv0.y, aself * sv0.z, aself * sv0.w,
                     aself * sv1.x, aself * sv1.y, aself * sv1.z, aself * sv1.w };
    #pragma unroll 4
    for (int k = 0; k < KNEI; ++k) {
      const float a = s_attn[node][k];
      const float* np = np_base + k * D;
      const float4 nv0 = *reinterpret_cast<const float4*>(np);
      const float4 nv1 = *reinterpret_cast<const float4*>(np + 4);
      acc[0] += a * nv0.x; acc[1] += a * nv0.y; acc[2] += a * nv0.z; acc[3] += a * nv0.w;
      acc[4] += a * nv1.x; acc[5] += a * nv1.y; acc[6] += a * nv1.z; acc[7] += a * nv1.w;
    }
    float* cdst = &s_ctx[node][seg * 8];
    #pragma unroll
    for (int i = 0; i < 8; ++i) cdst[i] = acc[i];
  }
  __syncthreads();

  // ---- phase 4: out = relu(ctx[16x128] @ W[128x128]) via WMMA ----------
  // 8 waves, wave w owns output columns [16w, 16w+16).
  // 32-bit A 16x4 layout: lanes 0-15 hold K=0(v0),K=1(v1); lanes 16-31 K=2,K=3.
  // B 4x16 mirrored: lanes 0-15 rows K=0/1, lanes 16-31 rows K=2/3.
  const int wave  = tid >> 5;
  const int lane  = tid & 31;
  const int lhalf = lane >> 4;   // selects K pair (A/B), M-half (C)
  const int l16   = lane & 15;   // M for A, N for B/C
  const int n0    = wave * 16;

  v8f c = {};
  #pragma unroll 4
  for (int kb = 0; kb < D / 4; ++kb) {
    const int k0 = kb * 4 + lhalf * 2;
    v2f a, b;
    a.x = s_ctx[l16][k0];
    a.y = s_ctx[l16][k0 + 1];
    b.x = weights[(size_t)(k0    ) * DOUT + n0 + l16];  // hot 64KB, cache-resident
    b.y = weights[(size_t)(k0 + 1) * DOUT + n0 + l16];
    // D = A x B + C  (8 args: neg_a, A, neg_b, B, c_mod, C, reuse_a, reuse_b)
    c = __builtin_amdgcn_wmma_f32_16x16x4_f32(false, a, false, b, (short)0, c,
                                              false, false);
  }

  // C/D layout: VGPR j, lanes 0-15 -> row j, lanes 16-31 -> row j+8; col = l16.
  // Fast path (all tiles when N % 16 == 0): one base address, 8 stores with
  // immediate offsets -- no per-row exec-mask churn.
  if (base + TILE <= n_nodes) {
    float* op = out + (size_t)(base + lhalf * 8) * DOUT + n0 + l16;
    #pragma unroll
    for (int j = 0; j < 8; ++j)
      op[(size_t)j * DOUT] = fmaxf(c[j], 0.0f);
  } else {
    #pragma unroll
    for (int j = 0; j < 8; ++j) {
      const int m = j + lhalf * 8;
      if (base + m < n_nodes)
        out[(size_t)(base + m) * DOUT + n0 + l16] = fmaxf(c[j], 0.0f);
    }
  }
}

extern "C" void kernel_launch(void* const* d_in, const int* in_sizes, int n_in,
                              void* d_out, int out_size, void* d_ws, size_t ws_size,
                              hipStream_t stream) {
  const float* self_vecs  = (const float*)d_in[0];
  const float* neigh_vecs = (const float*)d_in[1];
  const float* weights    = (const float*)d_in[2];
  float* out = (float*)d_out;

  const int n_nodes = in_sizes[0] / D;             // 100000
  const int tiles   = (n_nodes + TILE - 1) / TILE; // 6250 workgroups

  gat_attn_wmma<<<tiles, 256, 0, stream>>>(self_vecs, neigh_vecs, weights, out,
                                           n_nodes);
  (void)n_in; (void)out_size; (void)d_ws; (void)ws_size;
}